// EigenHora_13606456394294
// MI455X (gfx1250) — compile-verified
//
#include <hip/hip_runtime.h>

typedef __attribute__((ext_vector_type(2))) float v2f;
typedef __attribute__((ext_vector_type(8))) float v8f;

#define KC 32            // K chunk staged in LDS
#define ASTRIDE 36       // 32 + 4 pad: conflict-free strided A-fragment reads (144B rows, 16B-aligned)
#define MBLK 128         // rows of flat jac per block (8 16-row tiles)
#define RF 256           // R*F (GEMM N dimension)
#define FDIM 64          // F
#define NCHUNK 160       // 128 jac1 i-tiles/b + 32 jac2 i-tiles/b
#define BATCH 128

// Low 32 bits of a generic pointer into __shared__ == LDS byte offset (ISA: LDS
// aperture truncates addr[31:0]).
__device__ __forceinline__ unsigned lds_addr32(const void* p) {
    return (unsigned)(unsigned long long)p;
}

// Async global -> LDS copy of 16 bytes (CDNA5, tracked by ASYNCcnt).
__device__ __forceinline__ void async_copy_b128(unsigned lds_byte, const void* gptr) {
    asm volatile("global_load_async_to_lds_b128 %0, %1, off"
                 :: "v"(lds_byte), "v"(gptr)
                 : "memory");
}

__device__ __forceinline__ void wait_asynccnt0() {
    asm volatile("s_wait_asynccnt 0x0" ::: "memory");
}

// Flat-GEMM view: jac is [Mtot = B*Di, Dj]; compute P = jac @ Ub_flat[Dj, 256]
// fused with  partial[tile][f] = sum_{i in tile, r} Ua[i,r,f] * P[i, r*64+f].
//
// Block = 8 waves, 128 rows x 256 cols, K chunked by 32 through double-buffered
// LDS filled with global_load_async_to_lds_b128 (stage of chunk n+1 overlaps
// WMMA compute of chunk n). Wave w: row-group rg = w>>2 (4 row-tiles),
// col-group c0 = w&3 owning col-tiles {c0, c0+4, c0+8, c0+12} (= all r for
// f-block c0*16) -> the Ua contraction is wave-local.
__global__ __launch_bounds__(256)
void hora_gemm_contract(const float* __restrict__ jac,
                        const float* __restrict__ Ua,   // [Di, R, F] (i-indexed)
                        const float* __restrict__ Ub,   // [Dj, R, F] (j-indexed)
                        float* __restrict__ ws,
                        int Dj, int tileShift, int chunkBase)
{
    __shared__ float ldsA[2][MBLK * ASTRIDE];   // 2 x 18 KB
    __shared__ float ldsB[2][KC * RF];          // 2 x 32 KB

    const int tid  = threadIdx.x;
    const int wave = tid >> 5;
    const int lane = tid & 31;
    const int lrow = lane & 15;              // A row / B-C column within 16
    const int lhi  = lane >> 4;              // 0: K={0,1}; 1: K={2,3}
    const int rg   = wave >> 2;              // row group (0..1)
    const int c0   = wave & 3;               // col group -> f-block c0*16

    const int rowBase = blockIdx.x * MBLK;   // flat row = b*Di + i

    // Issue the async stage of one 128x32 A chunk + 32x256 B chunk into buffer `buf`.
    auto stage_async = [&](int kc, int buf) {
        // A: 1024 float4, coalesced, 4 per thread
        for (int q = tid; q < 1024; q += 256) {
            const int row = q >> 3;                  // 8 float4 per row
            const int c4  = (q & 7) << 2;
            async_copy_b128(lds_addr32(&ldsA[buf][row * ASTRIDE + c4]),
                            jac + (size_t)(rowBase + row) * Dj + kc + c4);
        }
        // B: 2048 float4, coalesced, 8 per thread (L2-resident U factor)
        for (int q = tid; q < 2048; q += 256) {
            const int row = q >> 6;                  // 64 float4 per row
            const int c4  = (q & 63) << 2;
            async_copy_b128(lds_addr32(&ldsB[buf][row * RF + c4]),
                            Ub + (size_t)(kc + row) * RF + c4);
        }
    };

    v8f acc[4][4] = {};                      // [row-tile][r]

    const int nChunks = Dj / KC;
    stage_async(0, 0);
    wait_asynccnt0();
    __syncthreads();

    for (int ci = 0; ci < nChunks; ++ci) {
        const int buf = ci & 1;
        if (ci + 1 < nChunks)
            stage_async((ci + 1) * KC, buf ^ 1);   // overlap with compute below

        const float* __restrict__ lA = ldsA[buf];
        const float* __restrict__ lB = ldsB[buf];
        for (int ks = 0; ks < KC; ks += 4) {
            const int klo = ks + (lhi << 1);         // this lane-half's first K
            v2f a[4];
#pragma unroll
            for (int mt = 0; mt < 4; ++mt)
                a[mt] = *(const v2f*)(&lA[((rg << 6) + (mt << 4) + lrow) * ASTRIDE + klo]);
            v2f bf[4];
#pragma unroll
            for (int r = 0; r < 4; ++r) {
                const int c = (r << 6) + (c0 << 4) + lrow;
                bf[r].x = lB[klo * RF + c];
                bf[r].y = lB[(klo + 1) * RF + c];
            }
#pragma unroll
            for (int mt = 0; mt < 4; ++mt)
#pragma unroll
                for (int r = 0; r < 4; ++r)
                    acc[mt][r] = __builtin_amdgcn_wmma_f32_16x16x4_f32(
                        false, a[mt], false, bf[r], (short)0, acc[mt][r], false, false);
        }

        wait_asynccnt0();      // next chunk's LDS writes complete (per-wave counter)
        __syncthreads();       // and all waves done reading current buffers
    }

    // ---- contract with Ua over 16 rows and r, per f = c0*16 + lrow ----
    // C layout: VGPR v on lane L holds C[M = v + 8*(L/16), N = L%16].
    const int f = (c0 << 4) + lrow;
    const int tilesMask = (1 << tileShift) - 1;      // tilesPerB - 1
#pragma unroll
    for (int mt = 0; mt < 4; ++mt) {
        const int gt  = (rowBase >> 4) + (rg << 2) + mt;   // global 16-row tile id
        const int b   = gt >> tileShift;
        const int lt  = gt & tilesMask;                    // i-tile within b
        const int i0  = lt << 4;
        float s = 0.0f;
#pragma unroll
        for (int r = 0; r < 4; ++r) {
            const size_t ucol = (size_t)(r << 6) + f;
#pragma unroll
            for (int v = 0; v < 8; ++v) {
                const int i = i0 + v + (lhi << 3);
                s += Ua[(size_t)i * RF + ucol] * acc[mt][r][v];
            }
        }
        s += __shfl_xor(s, 16, 32);                  // merge M-halves
        if (lane < 16)
            ws[((size_t)b * NCHUNK + (size_t)(chunkBase + lt)) * FDIM + f] = s;
    }
}

// Deterministic final reduction over the 160 per-tile partials.
__global__ __launch_bounds__(256)
void hora_reduce(const float* __restrict__ ws, float* __restrict__ out)
{
    const int idx = blockIdx.x * blockDim.x + threadIdx.x;  // b*64 + f
    if (idx >= BATCH * FDIM) return;
    const int b = idx >> 6;
    const int f = idx & 63;
    float s = 0.0f;
    for (int c = 0; c < NCHUNK; ++c)
        s += ws[((size_t)b * NCHUNK + c) * FDIM + f];
    out[idx] = s;
}

extern "C" void kernel_launch(void* const* d_in, const int* in_sizes, int n_in,
                              void* d_out, int out_size, void* d_ws, size_t ws_size,
                              hipStream_t stream) {
    const float* jac1 = (const float*)d_in[0];  // [128, 2048, 512]
    const float* jac2 = (const float*)d_in[1];  // [128, 512, 2048]
    const float* U1a  = (const float*)d_in[2];  // [2048, 4, 64]
    const float* U1b  = (const float*)d_in[3];  // [512, 4, 64]
    const float* U2a  = (const float*)d_in[4];  // [512, 4, 64]
    const float* U2b  = (const float*)d_in[5];  // [2048, 4, 64]
    float* ws  = (float*)d_ws;                  // needs 128*160*64*4 = 5,242,880 B
    float* out = (float*)d_out;                 // [128, 64]

    (void)in_sizes; (void)n_in; (void)out_size; (void)ws_size;

    dim3 block(256);
    // jac1: Mtot = 128*2048 rows, K = 512; 128 i-tiles/b (shift 7), chunks [0,128)
    hora_gemm_contract<<<dim3((BATCH * 2048) / MBLK), block, 0, stream>>>(
        jac1, U1a, U1b, ws, 512, 7, 0);
    // jac2: Mtot = 128*512 rows, K = 2048; 32 i-tiles/b (shift 5), chunks [128,160)
    hora_gemm_contract<<<dim3((BATCH * 512) / MBLK), block, 0, stream>>>(
        jac2, U2a, U2b, ws, 2048, 5, 128);

    hora_reduce<<<(BATCH * FDIM + 255) / 256, 256, 0, stream>>>(ws, out);
}